// Block_65180423685630
// MI455X (gfx1250) — compile-verified
//
#include <hip/hip_runtime.h>
#include <math.h>

typedef __attribute__((ext_vector_type(2))) float v2f;
typedef __attribute__((ext_vector_type(8))) float v8f;

#define B_    16
#define NPIX  16384
#define WIMG  128
#define HIMG  128
#define CDIM  64
#define NHEAD 2
#define HDQ   32
#define NTOK  (B_ * NPIX)   // 262144
#define MLPH  256

// ---------------- device helpers ----------------
__device__ __forceinline__ float softplusf_(float x) {
  return (x > 20.f) ? x : log1pf(expf(x));
}
__device__ __forceinline__ float sigmoidf_(float x) {
  return 1.f / (1.f + expf(-x));
}
__device__ __forceinline__ float safe_powf_(float x, float p) {
  float xr = fmaxf(x, 0.f);
  return (xr > 0.f) ? powf(xr, p) : 0.f;
}

// STEPS k-steps (4 each) of V_WMMA_F32_16X16X4_F32.
// A: 16 x (4*STEPS) tile in LDS (row-major, pitch lda)
// B: (4*STEPS) x 16 tile in LDS (row-major, pitch ldb)
// Fragments are register-prefetched; WMMAs then issue back-to-back.
template <int STEPS>
__device__ __forceinline__ void wmma_chunk(const float* As, int lda,
                                           const float* Bs, int ldb,
                                           int lane, v8f& acc) {
  const int half = lane >> 4;   // 0: k0,k0+1  1: k0+2,k0+3  (ISA 7.12.2)
  const int m = lane & 15;      // row for A, col for B
  v2f af[STEPS], bf[STEPS];
#pragma unroll
  for (int s = 0; s < STEPS; ++s) {
    const int k0 = s * 4 + half * 2;
    af[s].x = As[m * lda + k0];
    af[s].y = As[m * lda + k0 + 1];
  }
#pragma unroll
  for (int s = 0; s < STEPS; ++s) {
    const int k0 = s * 4 + half * 2;
    bf[s].x = Bs[k0 * ldb + m];
    bf[s].y = Bs[(k0 + 1) * ldb + m];
  }
#pragma unroll
  for (int s = 0; s < STEPS; ++s)
    acc = __builtin_amdgcn_wmma_f32_16x16x4_f32(false, af[s], false, bf[s],
                                                (short)0, acc, false, false);
}

// ---------------- kernels ----------------
__global__ void zero_kernel(float* __restrict__ p, int n) {
  int i = blockIdx.x * 256 + threadIdx.x;
  if (i < n) p[i] = 0.f;
}

// Fused LN1 + h1 @ [w_qg | w_kv] (64 -> 256):
//   q -> /scale -> qsc (pre-pow, 64/token)
//   k -> +pos_enc -> /scale -> ksc (pre-pow, 64/token)
//   gate, v -> raw buffers
__global__ void qkvg_kernel(const float* __restrict__ x,
                            const float* __restrict__ ln1_g,
                            const float* __restrict__ ln1_b,
                            const float* __restrict__ w_qg,
                            const float* __restrict__ w_kv,
                            const float* __restrict__ pos_enc,
                            const float* __restrict__ scale_p,
                            float* __restrict__ qsc,
                            float* __restrict__ ksc,
                            float* __restrict__ gate,
                            float* __restrict__ vbuf) {
  __shared__ __align__(16) float At[16][68];
  __shared__ __align__(16) float Ws[64][272];   // [k][col 0..255], 2*272%64==32
  __shared__ float ps[16][16];
  __shared__ float pq[16][16];
  const int t = threadIdx.x;
  const long row0 = (long)blockIdx.x * 16;
  // ---- stage x tile + LN1 partial sums; stage weights concurrently ----
  const int r = t >> 4, c4 = (t & 15) * 4;
  const float4 xv = *(const float4*)&x[(row0 + r) * CDIM + c4];
  ps[r][t & 15] = xv.x + xv.y + xv.z + xv.w;
  pq[r][t & 15] = xv.x * xv.x + xv.y * xv.y + xv.z * xv.z + xv.w * xv.w;
  for (int i = t; i < 64 * 64; i += 256) {  // 64 rows x 64 float4 cols
    const int wr = i >> 6, wc4 = (i & 63) * 4;
    const float* src = (wc4 < 128) ? &w_qg[wr * 128 + wc4]
                                   : &w_kv[wr * 128 + (wc4 - 128)];
    *(float4*)&Ws[wr][wc4] = *(const float4*)src;
  }
  __syncthreads();
  float a = 0.f, a2 = 0.f;
#pragma unroll
  for (int i = 0; i < 16; ++i) { a += ps[r][i]; a2 += pq[r][i]; }
  const float mu = a * (1.f / 64.f);
  const float rstd = rsqrtf(a2 * (1.f / 64.f) - mu * mu + 1e-5f);
  const float4 gv = *(const float4*)&ln1_g[c4];
  const float4 bv = *(const float4*)&ln1_b[c4];
  float4 hv;
  hv.x = (xv.x - mu) * rstd * gv.x + bv.x;
  hv.y = (xv.y - mu) * rstd * gv.y + bv.y;
  hv.z = (xv.z - mu) * rstd * gv.z + bv.z;
  hv.w = (xv.w - mu) * rstd * gv.w + bv.w;
  *(float4*)&At[r][c4] = hv;
  __syncthreads();
  // ---- GEMM + epilogue ----
  const int wave = t >> 5, lane = t & 31;
  const int half = lane >> 4, n = lane & 15;
  for (int ct = wave; ct < 16; ct += 8) {
    v8f cfr = {};
    wmma_chunk<16>(&At[0][0], 68, &Ws[0][ct * 16], 272, lane, cfr);
    const int colG = ct * 16 + n;   // 0..255
    const int sect = colG >> 6;     // 0=q 1=gate 2=k 3=v  (uniform per wave)
    const int cc = colG & 63;
    if (sect == 0 || sect == 2) {
      const float sci = 1.f / softplusf_(scale_p[cc]);
      float* base = (sect == 0) ? qsc : ksc;
#pragma unroll
      for (int v = 0; v < 8; ++v) {
        const long row = row0 + v + half * 8;
        float y = cfr[v];
        if (sect == 2) y += pos_enc[(row & (NPIX - 1)) * CDIM + cc];
        base[row * CDIM + cc] = y * sci;
      }
    } else {
      float* base = (sect == 1) ? gate : vbuf;
#pragma unroll
      for (int v = 0; v < 8; ++v) {
        const long row = row0 + v + half * 8;
        base[row * CDIM + cc] = cfr[v];
      }
    }
  }
}

// Per-(b,h) reductions: k_mean[64], kv_sim[64][16], kv_opp[64][16].
// k_cat is expanded from pre-pow ksc on the fly during LDS staging.
__global__ void red_kernel(const float* __restrict__ ksc,
                           const float* __restrict__ vbuf,
                           const float* __restrict__ power_p,
                           float* __restrict__ red) {
  __shared__ __align__(16) float kS[16][64];
  __shared__ __align__(16) float vS[16][36];
  __shared__ float pwS[32];
  const int t = threadIdx.x;
  const int bh = blockIdx.y;
  const int b = bh >> 1, h = bh & 1;
  const long base = (long)b * NPIX + (long)blockIdx.x * 512;
  if (t < 32) pwS[t] = 1.f + 4.f * sigmoidf_(power_p[h * HDQ + t]);
  __syncthreads();
  const int d = t >> 2;             // 0..63
  const int e0 = (t & 3) * 4;       // 0,4,8,12
  float accM = 0.f;
  float accS[4] = {0.f, 0.f, 0.f, 0.f};
  float accO[4] = {0.f, 0.f, 0.f, 0.f};
  for (int tok0 = 0; tok0 < 512; tok0 += 16) {
    for (int i = t; i < 16 * 32; i += 256) {  // expand k -> [pow(k), pow(-k)]
      const int tl = i >> 5, c = i & 31;
      const float kr = ksc[(base + tok0 + tl) * CDIM + h * HDQ + c];
      const float p = pwS[c];
      kS[tl][c] = safe_powf_(kr, p);
      kS[tl][32 + c] = safe_powf_(-kr, p);
    }
    if (t < 128) {  // v tile: b128 loads
      const int tl = t >> 3, c4 = (t & 7) * 4;
      *(float4*)&vS[tl][c4] =
          *(const float4*)&vbuf[(base + tok0 + tl) * CDIM + h * HDQ + c4];
    }
    __syncthreads();
#pragma unroll 4
    for (int tl = 0; tl < 16; ++tl) {
      const float kd = kS[tl][d];
      accM += kd;
#pragma unroll
      for (int j = 0; j < 4; ++j) {
        accS[j] += kd * vS[tl][e0 + j];        // v1 = head dims 0..15
        accO[j] += kd * vS[tl][16 + e0 + j];   // v2 = head dims 16..31
      }
    }
    __syncthreads();
  }
  const float invn = 1.f / (float)NPIX;  // inv_sqrt_n * inv_sqrt_n
  float* rb = red + bh * 2112;
  if ((t & 3) == 0) atomicAdd(&rb[d], accM * invn);
#pragma unroll
  for (int j = 0; j < 4; ++j) {
    atomicAdd(&rb[64 + d * 16 + e0 + j], accS[j] * invn);
    atomicAdd(&rb[64 + 1024 + d * 16 + e0 + j], accO[j] * invn);
  }
}

// Per-token linear attention: expand q_sim from pre-pow qsc, z-normalize,
// q@kv matvecs -> xa.
__global__ void attn_kernel(const float* __restrict__ qsc,
                            const float* __restrict__ power_p,
                            const float* __restrict__ red,
                            float* __restrict__ xa) {
  __shared__ __align__(16) float R[2][2112];
  __shared__ float pw[2][32];
  const int t = threadIdx.x;
  const int b = blockIdx.y;
  if (t < 64) pw[t >> 5][t & 31] = 1.f + 4.f * sigmoidf_(power_p[t]);
  for (int i = t; i < 2 * 528; i += 256) {   // 1056 float4s
    const int hh = i / 528, o = (i - hh * 528) * 4;
    *(float4*)&R[hh][o] = *(const float4*)&red[(b * 2 + hh) * 2112 + o];
  }
  __syncthreads();
  const int tl = t >> 1, h = t & 1;
  const long token = (long)b * NPIX + (long)blockIdx.x * 128 + tl;
  float qr[32];
  const float* qp = qsc + token * CDIM + h * HDQ;
#pragma unroll
  for (int i = 0; i < 8; ++i)
    *(float4*)&qr[i * 4] = *(const float4*)&qp[i * 4];
  float qs[64];
#pragma unroll
  for (int i = 0; i < 32; ++i) {
    const float p = pw[h][i];
    qs[i] = safe_powf_(qr[i], p);        // q_pos
    qs[32 + i] = safe_powf_(-qr[i], p);  // q_neg
  }
  float zs = 0.f, zo = 0.f;
#pragma unroll
  for (int i = 0; i < 64; ++i) {
    zs += qs[i] * R[h][i];         // q_sim . k_mean
    zo += qs[i ^ 32] * R[h][i];    // q_opp . k_mean (halves swapped)
  }
  zs = 1.f / (zs + 1e-6f);
  zo = 1.f / (zo + 1e-6f);
  float xout[32];
  for (int e = 0; e < 16; ++e) {
    float xs = 0.f, xo = 0.f;
#pragma unroll
    for (int dd = 0; dd < 64; ++dd) {
      xs += qs[dd] * R[h][64 + dd * 16 + e];
      xo += qs[dd ^ 32] * R[h][1088 + dd * 16 + e];
    }
    xout[e] = xs * zs;
    xout[16 + e] = xo * zo;
  }
  float* orow = xa + token * CDIM + h * HDQ;
#pragma unroll
  for (int i = 0; i < 8; ++i)
    *(float4*)&orow[i * 4] = *(const float4*)&xout[i * 4];
}

// 5x5 depthwise conv (SAME), LDS-tiled: 16x16 pixels x 32 channels per block
// with 20x20 halo staged coalesced; output tile staged and written back b128.
__global__ void dwc_kernel(const float* __restrict__ vbuf,
                           const float* __restrict__ dwc_w,
                           const float* __restrict__ dwc_b,
                           float* __restrict__ dwc) {
  __shared__ __align__(16) float sV[20][20][36];
  __shared__ __align__(16) float sO[16][16][36];
  __shared__ float sW[32][25];
  const int t = threadIdx.x;
  const int bh = blockIdx.z;
  const int b = bh >> 1, h = bh & 1;
  const int ty = blockIdx.y * 16, tx = blockIdx.x * 16;
  const long basebc = (long)b * NPIX;
  for (int i = t; i < 32 * 25; i += 256) sW[i / 25][i % 25] = dwc_w[i];
  for (int i = t; i < 20 * 20 * 8; i += 256) {
    const int pix = i >> 3, c4 = (i & 7) * 4;
    const int iy = ty + pix / 20 - 2, ix = tx + pix % 20 - 2;
    float4 val = make_float4(0.f, 0.f, 0.f, 0.f);
    if (iy >= 0 && iy < HIMG && ix >= 0 && ix < WIMG)
      val = *(const float4*)&vbuf[(basebc + iy * WIMG + ix) * CDIM + h * HDQ + c4];
    *(float4*)&sV[pix / 20][pix % 20][c4] = val;
  }
  __syncthreads();
  const int py = t >> 4, px = t & 15;
  for (int c = 0; c < 32; ++c) {
    float acc = dwc_b[c];
#pragma unroll
    for (int ky = 0; ky < 5; ++ky)
#pragma unroll
      for (int kx = 0; kx < 5; ++kx)
        acc += sW[c][ky * 5 + kx] * sV[py + ky][px + kx][c];
    sO[py][px][c] = acc;
  }
  __syncthreads();
  for (int i = t; i < 16 * 16 * 8; i += 256) {
    const int pix = i >> 3, c4 = (i & 7) * 4;
    const int oy = ty + (pix >> 4), ox = tx + (pix & 15);
    *(float4*)&dwc[(basebc + oy * WIMG + ox) * CDIM + h * HDQ + c4] =
        *(const float4*)&sO[pix >> 4][pix & 15][c4];
  }
}

// u = (xa + v_dwc) * gate; x2 = u @ w_proj + b_proj + x (64->64 WMMA GEMM)
__global__ void proj_kernel(const float* __restrict__ xa,
                            const float* __restrict__ dwc,
                            const float* __restrict__ gate,
                            const float* __restrict__ w_proj,
                            const float* __restrict__ b_proj,
                            const float* __restrict__ xin,
                            float* __restrict__ x2) {
  __shared__ __align__(16) float U[32][68];
  __shared__ __align__(16) float Ws[64][80];
  const int t = threadIdx.x;
  const long row0 = (long)blockIdx.x * 32;
  __builtin_prefetch(&xin[(row0 + (t >> 3)) * CDIM], 0, 3);
  for (int i = t; i < 32 * 16; i += 256) {
    const int r = i >> 4, c4 = (i & 15) * 4;
    const long row = row0 + r;
    const float4 va = *(const float4*)&xa[row * CDIM + c4];
    const float4 vd = *(const float4*)&dwc[row * CDIM + c4];
    const float4 vg = *(const float4*)&gate[row * CDIM + c4];
    float4 u;
    u.x = (va.x + vd.x) * vg.x; u.y = (va.y + vd.y) * vg.y;
    u.z = (va.z + vd.z) * vg.z; u.w = (va.w + vd.w) * vg.w;
    *(float4*)&U[r][c4] = u;
  }
  for (int i = t; i < 64 * 16; i += 256) {
    const int r = i >> 4, c4 = (i & 15) * 4;
    *(float4*)&Ws[r][c4] = *(const float4*)&w_proj[r * CDIM + c4];
  }
  __syncthreads();
  const int wave = t >> 5, lane = t & 31;
  const int rt = wave >> 2, ct = wave & 3;   // 2 row-tiles x 4 col-tiles
  v8f cfr = {};
  wmma_chunk<16>(&U[rt * 16][0], 68, &Ws[0][ct * 16], 80, lane, cfr);
  const int half = lane >> 4, n = lane & 15;
  const int colG = ct * 16 + n;
#pragma unroll
  for (int v = 0; v < 8; ++v) {
    const long row = row0 + rt * 16 + v + half * 8;
    x2[row * CDIM + colG] = cfr[v] + b_proj[colG] + xin[row * CDIM + colG];
  }
}

// Fused LN2 + fc1 + GELU + fc2 + residual (one pass over x2 -> out).
// GEMM1: 16x256, K=64 (8 waves x 2 col-tiles). Mid kept in LDS.
// GEMM2: 16x64, K=256, K-split across wave pairs; partials joined via LDS.
__global__ void mlp_kernel(const float* __restrict__ x2,
                           const float* __restrict__ ln2_g,
                           const float* __restrict__ ln2_b,
                           const float* __restrict__ w_fc1,
                           const float* __restrict__ b_fc1,
                           const float* __restrict__ w_fc2,
                           const float* __restrict__ b_fc2,
                           float* __restrict__ out) {
  __shared__ __align__(16) float At[16][68];     // h2 tile
  __shared__ __align__(16) float Xr[16][68];     // raw x2 tile (residual)
  __shared__ __align__(16) float Ws1[64][264];   // w_fc1, 2*264%64==16
  __shared__ __align__(16) float Ws2[256][72];   // w_fc2, 2*72%64==16
  __shared__ __align__(16) float Mid[16][260];   // gelu outputs
  __shared__ float Part[4][16][17];              // K-split partial sums
  __shared__ float ps[16][16];
  __shared__ float pq[16][16];
  const int t = threadIdx.x;
  const long row0 = (long)blockIdx.x * 16;
  // ---- stage x2 tile + LN2 partials; stage both weight matrices ----
  const int r = t >> 4, c4 = (t & 15) * 4;
  const float4 xv = *(const float4*)&x2[(row0 + r) * CDIM + c4];
  *(float4*)&Xr[r][c4] = xv;
  ps[r][t & 15] = xv.x + xv.y + xv.z + xv.w;
  pq[r][t & 15] = xv.x * xv.x + xv.y * xv.y + xv.z * xv.z + xv.w * xv.w;
  for (int i = t; i < 64 * 64; i += 256) {
    const int wr = i >> 6, wc4 = (i & 63) * 4;
    *(float4*)&Ws1[wr][wc4] = *(const float4*)&w_fc1[wr * MLPH + wc4];
  }
  for (int i = t; i < 256 * 16; i += 256) {
    const int wr = i >> 4, wc4 = (i & 15) * 4;
    *(float4*)&Ws2[wr][wc4] = *(const float4*)&w_fc2[wr * CDIM + wc4];
  }
  __syncthreads();
  float a = 0.f, a2 = 0.f;
#pragma unroll
  for (int i = 0; i < 16; ++i) { a += ps[r][i]; a2 += pq[r][i]; }
  const float mu = a * (1.f / 64.f);
  const float rstd = rsqrtf(a2 * (1.f / 64.f) - mu * mu + 1e-5f);
  const float4 gv = *(const float4*)&ln2_g[c4];
  const float4 bv = *(const float4*)&ln2_b[c4];
  float4 hv;
  hv.x = (xv.x - mu) * rstd * gv.x + bv.x;
  hv.y = (xv.y - mu) * rstd * gv.y + bv.y;
  hv.z = (xv.z - mu) * rstd * gv.z + bv.z;
  hv.w = (xv.w - mu) * rstd * gv.w + bv.w;
  *(float4*)&At[r][c4] = hv;
  __syncthreads();
  const int wave = t >> 5, lane = t & 31;
  const int half = lane >> 4, n = lane & 15;
  // ---- GEMM1: h2 @ w_fc1 -> bias + GELU -> Mid ----
  for (int ct = wave; ct < 16; ct += 8) {
    v8f cfr = {};
    wmma_chunk<16>(&At[0][0], 68, &Ws1[0][ct * 16], 264, lane, cfr);
    const int colG = ct * 16 + n;
#pragma unroll
    for (int v = 0; v < 8; ++v) {
      float y = cfr[v] + b_fc1[colG];
      y = 0.5f * y * (1.f + erff(y * 0.70710678118654752440f));
      Mid[v + half * 8][colG] = y;
    }
  }
  __syncthreads();
  // ---- GEMM2: Mid @ w_fc2, K split across wave pairs ----
  const int ct2 = wave >> 1, kh = wave & 1;
  v8f acc = {};
  wmma_chunk<16>(&Mid[0][kh * 128], 260, &Ws2[kh * 128][ct2 * 16], 72, lane, acc);
  wmma_chunk<16>(&Mid[0][kh * 128 + 64], 260, &Ws2[kh * 128 + 64][ct2 * 16], 72,
                 lane, acc);
  if (kh == 1) {
#pragma unroll
    for (int v = 0; v < 8; ++v) Part[ct2][v + half * 8][n] = acc[v];
  }
  __syncthreads();
  if (kh == 0) {
    const int colG = ct2 * 16 + n;
#pragma unroll
    for (int v = 0; v < 8; ++v) {
      const int lr = v + half * 8;
      out[(row0 + lr) * CDIM + colG] =
          acc[v] + Part[ct2][lr][n] + b_fc2[colG] + Xr[lr][colG];
    }
  }
}

// ---------------- launcher ----------------
extern "C" void kernel_launch(void* const* d_in, const int* in_sizes, int n_in,
                              void* d_out, int out_size, void* d_ws, size_t ws_size,
                              hipStream_t stream) {
  const float* x       = (const float*)d_in[0];
  // d_in[1]=H, d_in[2]=W (ints; shapes are compile-time constants here)
  const float* ln1_g   = (const float*)d_in[3];
  const float* ln1_b   = (const float*)d_in[4];
  const float* w_qg    = (const float*)d_in[5];
  const float* w_kv    = (const float*)d_in[6];
  const float* pos_enc = (const float*)d_in[7];
  const float* scale_p = (const float*)d_in[8];
  const float* power_p = (const float*)d_in[9];
  const float* dwc_w   = (const float*)d_in[10];
  const float* dwc_b   = (const float*)d_in[11];
  const float* w_proj  = (const float*)d_in[12];
  const float* b_proj  = (const float*)d_in[13];
  const float* ln2_g   = (const float*)d_in[14];
  const float* ln2_b   = (const float*)d_in[15];
  const float* w_fc1   = (const float*)d_in[16];
  const float* b_fc1   = (const float*)d_in[17];
  const float* w_fc2   = (const float*)d_in[18];
  const float* b_fc2   = (const float*)d_in[19];

  float* ws = (float*)d_ws;
  const size_t nt = (size_t)NTOK;
  float* qsc  = ws;                   // nt*64 (pre-pow q, scaled)
  float* ksc  = qsc  + nt * 64;       // nt*64 (pre-pow k, +pos, scaled)
  float* gate = ksc  + nt * 64;       // nt*64
  float* vbuf = gate + nt * 64;       // nt*64
  float* xa   = vbuf + nt * 64;       // nt*64
  float* dwc  = xa   + nt * 64;       // nt*64
  float* x2   = dwc  + nt * 64;       // nt*64
  float* redb = x2   + nt * 64;       // 32*2112

  dim3 blk(256);
  zero_kernel<<<(32 * 2112 + 255) / 256, blk, 0, stream>>>(redb, 32 * 2112);
  qkvg_kernel<<<NTOK / 16, blk, 0, stream>>>(x, ln1_g, ln1_b, w_qg, w_kv,
                                             pos_enc, scale_p, qsc, ksc, gate,
                                             vbuf);
  red_kernel<<<dim3(NPIX / 512, B_ * NHEAD), blk, 0, stream>>>(ksc, vbuf,
                                                               power_p, redb);
  attn_kernel<<<dim3(NPIX / 128, B_), blk, 0, stream>>>(qsc, power_p, redb, xa);
  dwc_kernel<<<dim3(WIMG / 16, HIMG / 16, B_ * NHEAD), blk, 0, stream>>>(
      vbuf, dwc_w, dwc_b, dwc);
  proj_kernel<<<NTOK / 32, blk, 0, stream>>>(xa, dwc, gate, w_proj, b_proj, x, x2);
  mlp_kernel<<<NTOK / 16, blk, 0, stream>>>(x2, ln2_g, ln2_b, w_fc1, b_fc1,
                                            w_fc2, b_fc2, (float*)d_out);
}